// PCN_Decoder_31980326486624
// MI455X (gfx1250) — compile-verified
//
#include <hip/hip_runtime.h>
#include <hip/hip_bf16.h>

// MI455X / gfx1250, wave32. FP32 WMMA path (V_WMMA_F32_16X16X4_F32).

typedef float v2f __attribute__((ext_vector_type(2)));
typedef float v4f __attribute__((ext_vector_type(4)));
typedef float v8f __attribute__((ext_vector_type(8)));

#define EPSF 1e-5f

static __device__ __forceinline__ v8f wmma_f32x4(v2f a, v2f b, v8f c) {
    // (neg_a, A, neg_b, B, c_mod, C, reuse_a, reuse_b)
    return __builtin_amdgcn_wmma_f32_16x16x4_f32(false, a, false, b, (short)0, c, false, false);
}

// ---------------------------------------------------------------------------
// Small-M GEMM: Out[8,N] = A[8,K] @ B[N,K]^T + bias, optional ReLU.
// One wave per 16-column tile; M padded 8->16 via a 0/1 mask (branch-free).
// Software-pipelined: operands for step k+4 are loaded before the wmma of k.
// ---------------------------------------------------------------------------
__global__ __launch_bounds__(32)
void gemm8_wmma(const float* __restrict__ A, int lda,
                const float* __restrict__ Bm, int ldb,
                const float* __restrict__ bias,
                float* __restrict__ Out, int ldo,
                int K, int relu)
{
    const int lane = threadIdx.x;
    const int half = lane >> 4;
    const int l    = lane & 15;
    const int n    = blockIdx.x * 16 + l;

    const float msk = (l < 8) ? 1.0f : 0.0f;
    const float* __restrict__ arow = A + (l & 7) * lda + half * 2;
    const float* __restrict__ brow = Bm + n * ldb + half * 2;

    v8f acc = {};
    v2f a, bv;
    a.x  = msk * arow[0];
    a.y  = msk * arow[1];
    bv.x = brow[0];
    bv.y = brow[1];
    for (int k = 0; k < K - 4; k += 4) {
        v2f an, bn_;
        an.x  = msk * arow[k + 4];
        an.y  = msk * arow[k + 5];
        bn_.x = brow[k + 4];
        bn_.y = brow[k + 5];
        acc = wmma_f32x4(a, bv, acc);
        a = an; bv = bn_;
    }
    acc = wmma_f32x4(a, bv, acc);

    if (half == 0) {  // rows 0..7 live in lanes 0..15, VGPRs 0..7
        const float bs = bias[n];
        #pragma unroll
        for (int r = 0; r < 8; ++r) {
            float v = acc[r] + bs;
            if (relu) v = fmaxf(v, 0.0f);
            Out[r * ldo + n] = v;
        }
    }
}

// ---------------------------------------------------------------------------
// gridtab[g][o] = lin[g/4]*Wf1[o][0] + lin[g%4]*Wf1[o][1], g in [0,16)
// ---------------------------------------------------------------------------
__global__ __launch_bounds__(256)
void gridtab_kernel(const float* __restrict__ Wf1, float* __restrict__ gt)
{
    const int i = blockIdx.x * 256 + threadIdx.x;   // 16*512 = 8192 total
    if (i >= 16 * 512) return;
    const int g = i >> 9, o = i & 511;
    const float step = 0.1f / 3.0f;
    const float gx = -0.05f + (float)(g >> 2) * step;
    const float gy = -0.05f + (float)(g & 3) * step;
    gt[i] = gx * Wf1[o * 1029 + 0] + gy * Wf1[o * 1029 + 1];
}

// ---------------------------------------------------------------------------
// Fused folding pipeline: 32 points per block, one batch per blockIdx.y.
// Phase 1: y1 = relu(bn1(base1 + gridtab + coarse·Wf1[:,2:5]))  -> LDS
//          (rank-5 update replaces the naive K=1029 GEMM; 138 GFLOP -> ~0.7)
// Phase 2: y2 = relu(bn2(y1 @ Wf2^T + bf2)) via fp32 WMMA.
//          2 M-subtiles x 2 N-tiles per wave: 2 ds_load_b64 + 2 global_b64
//          feed 4 wmma per k-step (1.0 mem-op/wmma), 1-deep SW pipeline.
// Phase 3: fine = y2 @ Wo^T + bo + center (reference's buggy-reshape center)
// ---------------------------------------------------------------------------
__global__ __launch_bounds__(256)
void fold_fused(const float* __restrict__ coarse,   // [8][1024][3] (in d_out)
                const float* __restrict__ base1,    // [8][512]
                const float* __restrict__ gridtab,  // [16][512]
                const float* __restrict__ Wf1,      // [512][1029]
                const float* __restrict__ g1, const float* __restrict__ be1,
                const float* __restrict__ m1, const float* __restrict__ v1,
                const float* __restrict__ Wf2,      // [512][512]
                const float* __restrict__ bf2,
                const float* __restrict__ g2, const float* __restrict__ be2,
                const float* __restrict__ m2, const float* __restrict__ v2,
                const float* __restrict__ Wo,       // [3][512]
                const float* __restrict__ bo,       // [3]
                float* __restrict__ fine)           // [8][16384][3]
{
    constexpr int Y2S = 516;  // y2 row stride: 8*516 mod 64 == 32 -> half-waves
                              // land in disjoint bank groups on store
    __shared__ float y1q[128 * 128];      // 64 KB: [kquad][point*4 + kelem]
    __shared__ float y2s[32 * Y2S];       // ~66 KB
    __shared__ float red[256 * 3];        //   3 KB

    const int t  = threadIdx.x;
    const int b  = blockIdx.y;
    const int m0 = blockIdx.x * 32;

    // ---- Phase 1: build y1 tile ----
    for (int i = 0; i < 16; ++i) {
        const int idx = i * 256 + t;
        const int q = idx >> 5;       // channel quad (uniform across a wave)
        const int p = idx & 31;       // point within tile (== lane)
        const int ng = m0 + p;
        const int j = ng >> 4, g = ng & 15;
        const float* cp = coarse + (b * 1024 + j) * 3;
        const float cx = cp[0], cy = cp[1], cz = cp[2];
        const v4f bse = *(const v4f*)(base1 + b * 512 + q * 4);
        const v4f gtv = *(const v4f*)(gridtab + g * 512 + q * 4);
        const v4f gv  = *(const v4f*)(g1 + q * 4);
        const v4f vv  = *(const v4f*)(v1 + q * 4);
        const v4f mv  = *(const v4f*)(m1 + q * 4);
        const v4f bev = *(const v4f*)(be1 + q * 4);
        v4f r;
        #pragma unroll
        for (int c = 0; c < 4; ++c) {
            const int o = q * 4 + c;
            float val = bse[c] + gtv[c]
                      + cx * Wf1[o * 1029 + 2]
                      + cy * Wf1[o * 1029 + 3]
                      + cz * Wf1[o * 1029 + 4];
            const float s = gv[c] * rsqrtf(vv[c] + EPSF);
            val = (val - mv[c]) * s + bev[c];
            r[c] = fmaxf(val, 0.0f);
        }
        *(v4f*)&y1q[q * 128 + p * 4] = r;   // b128 store
    }
    __syncthreads();

    // ---- Phase 2: y2 = relu(bn2(y1 @ Wf2^T + bf2)) with fp32 WMMA ----
    {
        const int w    = t >> 5;
        const int lane = t & 31;
        const int half = lane >> 4;
        const int l    = lane & 15;
        const int aoff0 = l * 4 + half * 2;          // M-subtile 0 (rows 0..15)
        const int aoff1 = (16 + l) * 4 + half * 2;   // M-subtile 1 (rows 16..31)
        for (int nt2 = 0; nt2 < 2; ++nt2) {
            const int n0 = (w * 4 + nt2 * 2) * 16 + l;   // N-tile pair n0, n0+16
            const int n1 = n0 + 16;
            const float* __restrict__ bp0 = Wf2 + n0 * 512 + half * 2;
            const float* __restrict__ bp1 = Wf2 + n1 * 512 + half * 2;
            v8f acc00 = {}, acc01 = {}, acc10 = {}, acc11 = {};
            v2f a0 = *(const v2f*)&y1q[aoff0];
            v2f a1 = *(const v2f*)&y1q[aoff1];
            v2f b0 = *(const v2f*)bp0;
            v2f b1 = *(const v2f*)bp1;
            for (int kq = 0; kq < 127; ++kq) {
                const v2f a0n = *(const v2f*)&y1q[(kq + 1) * 128 + aoff0];
                const v2f a1n = *(const v2f*)&y1q[(kq + 1) * 128 + aoff1];
                const v2f b0n = *(const v2f*)(bp0 + (kq + 1) * 4);
                const v2f b1n = *(const v2f*)(bp1 + (kq + 1) * 4);
                acc00 = wmma_f32x4(a0, b0, acc00);
                acc01 = wmma_f32x4(a0, b1, acc01);
                acc10 = wmma_f32x4(a1, b0, acc10);
                acc11 = wmma_f32x4(a1, b1, acc11);
                a0 = a0n; a1 = a1n; b0 = b0n; b1 = b1n;
            }
            acc00 = wmma_f32x4(a0, b0, acc00);
            acc01 = wmma_f32x4(a0, b1, acc01);
            acc10 = wmma_f32x4(a1, b0, acc10);
            acc11 = wmma_f32x4(a1, b1, acc11);

            const float s20 = g2[n0] * rsqrtf(v2[n0] + EPSF);
            const float s21 = g2[n1] * rsqrtf(v2[n1] + EPSF);
            const float mm0 = m2[n0], bt0 = be2[n0], bf0 = bf2[n0];
            const float mm1 = m2[n1], bt1 = be2[n1], bf1v = bf2[n1];
            #pragma unroll
            for (int r = 0; r < 8; ++r) {
                const int M = r + half * 8;
                y2s[M * Y2S + n0]        = fmaxf((acc00[r] + bf0 - mm0) * s20 + bt0, 0.0f);
                y2s[M * Y2S + n1]        = fmaxf((acc01[r] + bf1v - mm1) * s21 + bt1, 0.0f);
                y2s[(16 + M) * Y2S + n0] = fmaxf((acc10[r] + bf0 - mm0) * s20 + bt0, 0.0f);
                y2s[(16 + M) * Y2S + n1] = fmaxf((acc11[r] + bf1v - mm1) * s21 + bt1, 0.0f);
            }
        }
    }
    __syncthreads();

    // ---- Phase 3: fine = y2 @ Wo^T + bo + center ----
    {
        const int p = t >> 3;
        const int s = t & 7;
        float p0 = 0.0f, p1 = 0.0f, p2 = 0.0f;
        for (int o = s; o < 512; o += 8) {
            const float y = y2s[p * Y2S + o];
            p0 += y * Wo[o];
            p1 += y * Wo[512 + o];
            p2 += y * Wo[1024 + o];
        }
        red[t * 3 + 0] = p0;
        red[t * 3 + 1] = p1;
        red[t * 3 + 2] = p2;
    }
    __syncthreads();
    if (t < 96) {
        const int p = t / 3, c = t % 3;
        float sum = 0.0f;
        #pragma unroll
        for (int s = 0; s < 8; ++s) sum += red[(p * 8 + s) * 3 + c];
        const int ng = m0 + p;
        // center per the reference's buggy reshape [B,N,3] -> [B,3,N] -> T
        const int i  = c * 16384 + ng;
        const int r3 = i / 3;
        const int cc = i - r3 * 3;
        const float center = coarse[(b * 1024 + (r3 >> 4)) * 3 + cc];
        fine[(b * 16384 + ng) * 3 + c] = sum + bo[c] + center;
    }
}

// ---------------------------------------------------------------------------
extern "C" void kernel_launch(void* const* d_in, const int* in_sizes, int n_in,
                              void* d_out, int out_size, void* d_ws, size_t ws_size,
                              hipStream_t stream)
{
    const float* features = (const float*)d_in[0];
    const float* W1  = (const float*)d_in[1];
    const float* b1  = (const float*)d_in[2];
    const float* W2  = (const float*)d_in[3];
    const float* b2  = (const float*)d_in[4];
    const float* Wc  = (const float*)d_in[5];
    const float* bc  = (const float*)d_in[6];
    const float* Wf1 = (const float*)d_in[7];
    const float* bf1 = (const float*)d_in[8];
    const float* g1  = (const float*)d_in[9];
    const float* be1 = (const float*)d_in[10];
    const float* m1  = (const float*)d_in[11];
    const float* v1  = (const float*)d_in[12];
    const float* Wf2 = (const float*)d_in[13];
    const float* bf2 = (const float*)d_in[14];
    const float* g2  = (const float*)d_in[15];
    const float* be2 = (const float*)d_in[16];
    const float* m2  = (const float*)d_in[17];
    const float* v2  = (const float*)d_in[18];
    const float* Wo  = (const float*)d_in[19];
    const float* bo  = (const float*)d_in[20];

    float* out    = (float*)d_out;
    float* coarse = out;                 // 8*1024*3 = 24576 floats
    float* fine   = out + 8 * 1024 * 3;  // 8*16384*3 floats

    float* ws    = (float*)d_ws;
    float* h1    = ws;            // 8192
    float* h2    = ws + 8192;     // 8192
    float* base1 = ws + 16384;    // 4096 (16B-aligned offset)
    float* gt    = ws + 20480;    // 8192 (16B-aligned offset)

    // Coarse decoder (M=8 GEMMs, fp32 WMMA)
    gemm8_wmma<<<64,  32, 0, stream>>>(features, 1024, W1, 1024, b1, h1,     1024, 1024, 1);
    gemm8_wmma<<<64,  32, 0, stream>>>(h1,       1024, W2, 1024, b2, h2,     1024, 1024, 1);
    gemm8_wmma<<<192, 32, 0, stream>>>(h2,       1024, Wc, 1024, bc, coarse, 3072, 1024, 0);
    // base1 = features @ Wf1[:,5:]^T + bf1   (the broadcast part of GEMM1)
    gemm8_wmma<<<32,  32, 0, stream>>>(features, 1024, Wf1 + 5, 1029, bf1, base1, 512, 1024, 0);
    gridtab_kernel<<<32, 256, 0, stream>>>(Wf1, gt);

    dim3 gridF(16384 / 32, 8);
    fold_fused<<<gridF, 256, 0, stream>>>(coarse, base1, gt, Wf1,
                                          g1, be1, m1, v1,
                                          Wf2, bf2, g2, be2, m2, v2,
                                          Wo, bo, fine);
}